// BilinearLTN_85693187489909
// MI455X (gfx1250) — compile-verified
//
#include <hip/hip_runtime.h>
#include <hip/hip_bf16.h>

// BilinearLTN: scores[b] = h_emb[b]^T * W[r[b]] * t_emb[b]
// DIM=256, B=4096, N_REL=500 (whole rel table ~128MB -> fits MI455X 192MB L2).
// Memory-bound: ~1GB of W reads worst case. CDNA5 path:
//   TDM tensor_load_to_lds (double-buffered, hw row-padding) -> LDS
//   V_WMMA_F32_16X16X4_F32 chain for u = h^T W, then u . t reduction.

#define DIM   256
#define TILE  16
#define NPANEL (DIM / TILE)      // 16 row panels
#define ROWSTRIDE (DIM + 4)      // padded LDS row stride (floats); TDM pads 4 DW / 256 DW
#define PANEL_FLOATS (TILE * ROWSTRIDE)

typedef float v2f __attribute__((ext_vector_type(2)));
typedef float v8f __attribute__((ext_vector_type(8)));
typedef unsigned int u32x4 __attribute__((ext_vector_type(4)));
typedef int i32x4 __attribute__((ext_vector_type(4)));
typedef int i32x8 __attribute__((ext_vector_type(8)));

// Issue a TDM load of one 16x256 f32 panel (rows of W) into LDS at lds_ptr,
// with hardware padding of 4 DWORDs after every 256 DWORDs (row stride 260).
__device__ __forceinline__ void tdm_load_panel(const float* gsrc, const void* lds_ptr) {
    unsigned ga_lo = (unsigned)__builtin_amdgcn_readfirstlane((int)(unsigned)(uintptr_t)gsrc);
    unsigned ga_hi = (unsigned)__builtin_amdgcn_readfirstlane((int)(unsigned)((uintptr_t)gsrc >> 32));
    unsigned lds   = (unsigned)__builtin_amdgcn_readfirstlane((int)(unsigned)(uintptr_t)lds_ptr);

    u32x4 g0;
    g0[0] = 1u;                                        // count=1, user descriptor
    g0[1] = lds;                                       // lds_addr (byte offset)
    g0[2] = ga_lo;                                     // global_addr[31:0]
    g0[3] = (ga_hi & 0x01FFFFFFu) | 0x80000000u;       // global_addr[56:32] | type=2

    i32x8 g1 = {};
    // data_size=4B (2<<16) | pad_enable (1<<20) | pad_interval=256DW (7<<22)
    // | pad_amount=4DW (3<<25); workgroup_mask=0 (not in cluster)
    g1[0] = (int)((2u << 16) | (1u << 20) | (7u << 22) | (3u << 25));
    g1[1] = (int)(256u << 16);   // tensor_dim0 = 256  (bits [79:48], low half)
    g1[2] = (int)(256u << 16);   // tensor_dim1 = 256  (bits [95:80])
    g1[3] = (int)(256u << 16);   // tile_dim0   = 256  (bits [127:112])
    g1[4] = 16;                  // tile_dim1   = 16   (bits [143:128])
    g1[5] = 256;                 // tensor_dim0_stride = 256 (bits [207:160], low)
    g1[6] = 0;
    g1[7] = 0;

    i32x4 z4 = {};
    i32x8 z8 = {};
    __builtin_amdgcn_tensor_load_to_lds(g0, g1, z4, z4, z8, 0);
}

__global__ __launch_bounds__(256, 2)
void BilinearLTN_kernel(const int* __restrict__ hIdx,
                        const int* __restrict__ rIdx,
                        const int* __restrict__ tIdx,
                        const float* __restrict__ ent,
                        const float* __restrict__ rel,
                        float* __restrict__ out)
{
    __shared__ __attribute__((aligned(16))) float sh_h[DIM];
    __shared__ __attribute__((aligned(16))) float sh_t[DIM];
    __shared__ __attribute__((aligned(16))) float sh_w[2 * PANEL_FLOATS]; // double buffer
    __shared__ float sh_red[256];

    const int b    = blockIdx.x;
    const int tid  = threadIdx.x;
    const int lane = tid & 31;
    const int wave = tid >> 5;

    const float* __restrict__ hEmb = ent + (size_t)hIdx[b] * DIM;
    const float* __restrict__ tEmb = ent + (size_t)tIdx[b] * DIM;
    const float* __restrict__ W    = rel + (size_t)rIdx[b] * (size_t)(DIM * DIM);

    // Stage embeddings (256 threads, 256 floats each).
    sh_h[tid] = hEmb[tid];
    sh_t[tid] = tEmb[tid];

    // Prologue: kick off TDM for panel 0 into buffer 0.
    if (wave == 0) {
        tdm_load_panel(W, &sh_w[0]);
    }

    // Each of 8 waves owns two 16-column tiles of u = h^T W.
    const int tj0 = wave * 2;
    const int tj1 = wave * 2 + 1;

    v8f acc0 = {};
    v8f acc1 = {};

    const int colSel = lane & 15;              // N index within tile
    const int kSel   = (lane < 16) ? 0 : 2;    // lanes 0-15: K=0/1, lanes 16-31: K=2/3

    for (int ti = 0; ti < NPANEL; ++ti) {
        // All waves done computing panel ti-1 -> buffer (ti+1)&1 is reusable.
        __syncthreads();

        if (wave == 0) {
            if (ti + 1 < NPANEL) {
                // Prefetch next panel via TDM while we compute this one.
                tdm_load_panel(W + (size_t)(ti + 1) * (TILE * DIM),
                               &sh_w[((ti + 1) & 1) * PANEL_FLOATS]);
                __builtin_amdgcn_s_wait_tensorcnt(1);  // panel ti has landed
            } else {
                __builtin_amdgcn_s_wait_tensorcnt(0);  // last panel landed
            }
        }
        __syncthreads();   // panel ti visible to all waves

        const float* __restrict__ wbuf = &sh_w[(ti & 1) * PANEL_FLOATS];

        // ---- 4 K-steps of V_WMMA_F32_16X16X4_F32 per tile ----
        #pragma unroll
        for (int s = 0; s < 4; ++s) {
            const int kBase = ti * TILE + s * 4;   // global row (i) base

            // A (16x4 f32): A[m,k] = h[kBase+k], identical for all m.
            float2 hv = *(const float2*)(&sh_h[kBase + kSel]);
            v2f a; a.x = hv.x; a.y = hv.y;

            // B (4x16 f32): B[k,n] = W[kBase+k, 16*tj + n]
            const int rB = (s * 4 + kSel) * ROWSTRIDE;

            const float* bp0 = &wbuf[rB + tj0 * TILE + colSel];
            v2f b0; b0.x = bp0[0]; b0.y = bp0[ROWSTRIDE];
            acc0 = __builtin_amdgcn_wmma_f32_16x16x4_f32(
                false, a, false, b0, (short)0, acc0, false, false);

            const float* bp1 = &wbuf[rB + tj1 * TILE + colSel];
            v2f b1; b1.x = bp1[0]; b1.y = bp1[ROWSTRIDE];
            acc1 = __builtin_amdgcn_wmma_f32_16x16x4_f32(
                false, a, false, b1, (short)0, acc1, false, false);
        }
    }

    // D rows are replicated (all A rows identical): lane n (<16) vgpr0 holds
    // u[tj0*16+n]; lane n (>=16) of acc1 vgpr0 holds u[tj1*16+(n-16)].
    float val;
    if (lane < 16) {
        val = acc0[0] * sh_t[tj0 * TILE + lane];
    } else {
        val = acc1[0] * sh_t[tj1 * TILE + (lane - 16)];
    }

    // Block tree reduction of 256 partials.
    sh_red[tid] = val;
    __syncthreads();
    #pragma unroll
    for (int off = 128; off > 0; off >>= 1) {
        if (tid < off) sh_red[tid] += sh_red[tid + off];
        __syncthreads();
    }
    if (tid == 0) out[b] = sh_red[0];
}

extern "C" void kernel_launch(void* const* d_in, const int* in_sizes, int n_in,
                              void* d_out, int out_size, void* d_ws, size_t ws_size,
                              hipStream_t stream) {
    (void)n_in; (void)d_ws; (void)ws_size;
    const int*   hIdx = (const int*)d_in[0];
    const int*   rIdx = (const int*)d_in[1];
    const int*   tIdx = (const int*)d_in[2];
    const float* ent  = (const float*)d_in[3];
    const float* rel  = (const float*)d_in[4];
    float*       out  = (float*)d_out;

    const int B = in_sizes[0];
    (void)out_size;

    BilinearLTN_kernel<<<B, 256, 0, stream>>>(hIdx, rIdx, tIdx, ent, rel, out);
}